// HessianLoss_24257975288508
// MI455X (gfx1250) — compile-verified
//
#include <hip/hip_runtime.h>

typedef __attribute__((ext_vector_type(2))) float v2f;
typedef __attribute__((ext_vector_type(8))) float v8f;

struct F3 { float x, y, z; };
__device__ __forceinline__ F3 fsub(F3 a, F3 b) { return {a.x-b.x, a.y-b.y, a.z-b.z}; }
__device__ __forceinline__ F3 fcross(F3 a, F3 b) {
  return {a.y*b.z - a.z*b.y, a.z*b.x - a.x*b.z, a.x*b.y - a.y*b.x};
}
__device__ __forceinline__ float fdot(F3 a, F3 b) { return a.x*b.x + a.y*b.y + a.z*b.z; }
__device__ __forceinline__ F3 fscale(F3 a, float s) { return {a.x*s, a.y*s, a.z*s}; }
__device__ __forceinline__ F3 ld3(const float* __restrict__ p, int i) {
  return {p[3*(size_t)i], p[3*(size_t)i+1], p[3*(size_t)i+2]};
}

// ---------------- zero the H accumulator (18 floats per vertex) ----------------
__global__ void zero_ws_kernel(float* __restrict__ p, long n) {
  long i = (long)blockIdx.x * blockDim.x + threadIdx.x;
  long stride = (long)gridDim.x * blockDim.x;
  long n4 = n >> 2;
  float4* p4 = (float4*)p;
  for (long k = i; k < n4; k += stride) p4[k] = make_float4(0.f, 0.f, 0.f, 0.f);
  if (i == 0) for (long k = n4 << 2; k < n; ++k) p[k] = 0.f;
}

// ---------------- per-face compute + scatter (atomic f32, L2-resident) --------
__global__ void face_scatter_kernel(const float* __restrict__ vs,
                                    const float* __restrict__ vt,
                                    const int* __restrict__ face,
                                    float* __restrict__ H, int F) {
  int f = blockIdx.x * blockDim.x + threadIdx.x;
  if (f >= F) return;
  int i0 = face[3*(size_t)f + 0];
  int i1 = face[3*(size_t)f + 1];
  int i2 = face[3*(size_t)f + 2];

  F3 V0 = ld3(vs, i0), V1 = ld3(vs, i1), V2 = ld3(vs, i2);
  F3 T0 = ld3(vt, i0), T1 = ld3(vt, i1), T2 = ld3(vt, i2);
  F3 D0 = fsub(T0, V0), D1 = fsub(T1, V1), D2 = fsub(T2, V2);

  F3 v21 = fsub(V2, V1), v02 = fsub(V0, V2), v10 = fsub(V1, V0);
  F3 nv  = fcross(v21, v02);
  float dblA = sqrtf(fdot(nv, nv));
  F3 u = fscale(nv, 1.0f / dblA);

  // eperp(v) = cross(u,v) * (|v| / (dblA*|cross(u,v)|))
  F3 c02 = fcross(u, v02);
  F3 e02 = fscale(c02, sqrtf(fdot(v02, v02)) / (dblA * sqrtf(fdot(c02, c02))));
  F3 c10 = fcross(u, v10);
  F3 e10 = fscale(c10, sqrtf(fdot(v10, v10)) / (dblA * sqrtf(fdot(c10, c10))));

  float area = 0.5f * dblA;
  F3 d10 = fsub(D1, D0), d20 = fsub(D2, D0);

  float e02a[3] = {e02.x, e02.y, e02.z};
  float e10a[3] = {e10.x, e10.y, e10.z};
  float d10a[3] = {d10.x, d10.y, d10.z};
  float d20a[3] = {d20.x, d20.y, d20.z};

  float aw[3][3];
#pragma unroll
  for (int a = 0; a < 3; ++a)
#pragma unroll
    for (int j = 0; j < 3; ++j)
      aw[a][j] = area * (e02a[a] * d10a[j] + e10a[a] * d20a[j]);

  // corner -> (vertex, coefficient) mapping:  i1<-e02, i2<-e10, i0<- -(e02+e10)
  int   vidx[3]    = {i1, i2, i0};
  float coefs[3][3] = {
    { e02a[0],            e02a[1],            e02a[2]            },
    { e10a[0],            e10a[1],            e10a[2]            },
    { -(e02a[0]+e10a[0]), -(e02a[1]+e10a[1]), -(e02a[2]+e10a[2]) }
  };
  // (a,b) pairs for Hxx,Hyy,Hzz,Hxy,Hxz,Hyz
  const int pa[6] = {0, 1, 2, 0, 0, 1};
  const int pb[6] = {0, 1, 2, 1, 2, 2};

#pragma unroll
  for (int c = 0; c < 3; ++c) {
    float* base = H + (size_t)vidx[c] * 18;   // 18 contiguous floats per corner
#pragma unroll
    for (int ab = 0; ab < 6; ++ab) {
      float ca = coefs[c][pa[ab]];
#pragma unroll
      for (int j = 0; j < 3; ++j)
        unsafeAtomicAdd(base + ab * 3 + j, ca * aw[pb[ab]][j]);
    }
  }
}

// -------------- weighted sum of squares: per-block partials -------------------
__global__ void sq_reduce_kernel(const float* __restrict__ H,
                                 float* __restrict__ partials, long total) {
  __shared__ float smem[256];
  long i = (long)blockIdx.x * blockDim.x + threadIdx.x;
  long stride = (long)gridDim.x * blockDim.x;
  float acc = 0.f;
  for (long k = i; k < total; k += stride) {
    float h = H[k];
    int ab = (int)((k / 3) % 6);         // layout [v][ab][j]
    float w = (ab < 3) ? 1.0f : 2.0f;    // off-diagonal terms doubled
    acc += w * h * h;
  }
  smem[threadIdx.x] = acc;
  __syncthreads();
  for (int off = 128; off > 0; off >>= 1) {
    if (threadIdx.x < off) smem[threadIdx.x] += smem[threadIdx.x + off];
    __syncthreads();
  }
  if (threadIdx.x == 0) partials[blockIdx.x] = smem[0];
}

// -------------- final reduce: one wave32, WMMA f32 reduction tree -------------
__global__ void final_reduce_kernel(const float* __restrict__ partials, int P,
                                    float* __restrict__ out, long n) {
  int lane = threadIdx.x;                 // 32 threads, all active -> EXEC all 1s
  float p = 0.f;
  for (int i = lane; i < P; i += 32) p += partials[i];

  // D = ones(16x4) * B(4x16) : every column of D = sum of the 4 B values in
  // that column; all 64 B slots cover the 32 lanes' {p, 0} exactly once.
  v2f a; a[0] = 1.0f; a[1] = 1.0f;        // A = ones
  v2f b; b[0] = p;    b[1] = 0.0f;        // B = per-lane partials
  v8f c = {};
  c = __builtin_amdgcn_wmma_f32_16x16x4_f32(
      /*neg_a=*/false, a, /*neg_b=*/false, b,
      /*c_mod=*/(short)0, c, /*reuse_a=*/false, /*reuse_b=*/false);
  float cs = c[0];                        // row 0 (lanes 0-15) / row 8 (16-31): column sums
  // each 16-lane half holds all 16 column sums -> xor-reduce within the half
  cs += __shfl_xor(cs, 8, 32);
  cs += __shfl_xor(cs, 4, 32);
  cs += __shfl_xor(cs, 2, 32);
  cs += __shfl_xor(cs, 1, 32);
  if (lane == 0) out[0] = cs / (3.0f * (float)n);
}

extern "C" void kernel_launch(void* const* d_in, const int* in_sizes, int n_in,
                              void* d_out, int out_size, void* d_ws, size_t ws_size,
                              hipStream_t stream) {
  const float* vs   = (const float*)d_in[0];
  const float* vt   = (const float*)d_in[1];
  const int*   face = (const int*)d_in[2];
  long n = (long)in_sizes[0] / 3;         // vertices
  int  F = in_sizes[2] / 3;               // faces

  float* H = (float*)d_ws;                // 18*n floats (75.5 MB for n=1M)
  long Hcount = 18 * n;
  float* partials = H + Hcount;           // RB floats
  const int RB = 1024;

  zero_ws_kernel<<<2048, 256, 0, stream>>>(H, Hcount);
  face_scatter_kernel<<<(F + 255) / 256, 256, 0, stream>>>(vs, vt, face, H, F);
  sq_reduce_kernel<<<RB, 256, 0, stream>>>(H, partials, Hcount);
  final_reduce_kernel<<<1, 32, 0, stream>>>(partials, RB, (float*)d_out, n);
}